// FrameRoPE_22282290331803
// MI455X (gfx1250) — compile-verified
//
#include <hip/hip_runtime.h>
#include <cstdint>

typedef float v4f __attribute__((ext_vector_type(4)));

#define NFRAMES 32
#define TOKENS  257            // tokens per frame (256 video + 1 audio)
#define PSTRIDE 16             // 15 active pairs + 1 pad (float2 each)
#define BH      64             // b*h = 4*16
#define D       64             // head dim
#define SEQ     (NFRAMES * TOKENS)          // 8224
#define TENSOR_ELEMS ((size_t)BH * SEQ * D) // 33,685,504 floats per tensor

// ---------------------------------------------------------------------------
// Kernel A: build cos/sin table from the freqs input (32,17,17,30).
// Table layout: tbl[(n*257 + tok)*16 + p] = {cos(f), sin(f)} for pair p<15,
// zero padding at p==15. One entry per (frame, token, rotation pair).
// ---------------------------------------------------------------------------
__global__ __launch_bounds__(256) void build_cs_table(
    const float* __restrict__ freqs, float2* __restrict__ tbl) {
  int idx = blockIdx.x * blockDim.x + threadIdx.x;
  const int TOT = NFRAMES * TOKENS * PSTRIDE;   // 131,584
  if (idx >= TOT) return;
  int p   = idx & 15;
  int rem = idx >> 4;              // n*257 + tok
  int tok = rem % TOKENS;
  int n   = rem / TOKENS;
  float2 cs = make_float2(0.f, 0.f);
  if (p < 15) {
    int pi = (tok < 256) ? (tok >> 4) : 16;   // audio token sits at (16,16)
    int pj = (tok < 256) ? (tok & 15) : 16;
    int c  = 2 * p;                           // freqs repeat pairwise
    float f = freqs[(((size_t)n * 17 + pi) * 17 + pj) * 30 + c];
    float s, co;
    __sincosf(f, &s, &co);
    cs = make_float2(co, s);
  }
  tbl[idx] = cs;
}

// ---------------------------------------------------------------------------
// Kernel B: apply RoPE. Block = (tensor q/k, frame n, chunk of 4 bh slices).
// Stages the frame's 33KB cos/sin slice into LDS via async global->LDS b128
// (ASYNCcnt), then streams tokens with nontemporal b128 loads/stores.
// ---------------------------------------------------------------------------
__global__ __launch_bounds__(256) void frame_rope_main(
    const float* __restrict__ q, const float* __restrict__ k,
    const float2* __restrict__ tbl, float* __restrict__ out) {
  __shared__ float2 cs[TOKENS * PSTRIDE];   // 257*16*8 = 32,896 bytes

  int bid   = blockIdx.x;
  int tsel  = bid & 1;            // 0 -> q, 1 -> k
  int n     = (bid >> 1) & 31;    // frame
  int chunk = bid >> 6;           // 0..15, 4 bh slices each
  const float* src = tsel ? k : q;
  float*       dst = out + (size_t)tsel * TENSOR_ELEMS;

  // --- async copy of this frame's cos/sin table into LDS -------------------
  {
    uint64_t gbase = (uint64_t)(uintptr_t)(tbl + (size_t)n * TOKENS * PSTRIDE);
    unsigned lds_base = (unsigned)(uintptr_t)(&cs[0]);  // flat addr low 32 = LDS offset
    const int NCHUNK = (TOKENS * PSTRIDE) / 2;          // 2056 x 16B transfers
    for (int i = threadIdx.x; i < NCHUNK; i += blockDim.x) {
      unsigned voff = (unsigned)i * 16u;                // byte offset into slice
      unsigned laddr = lds_base + voff;
      asm volatile("global_load_async_to_lds_b128 %0, %1, %2"
                   :: "v"(laddr), "v"(voff), "s"(gbase)
                   : "memory");
    }
    asm volatile("s_wait_asynccnt 0x0" ::: "memory");
    __syncthreads();
  }

  // --- stream 4 (b,h) slices of this frame ---------------------------------
  const int SLOTS = TOKENS * 16;  // 16 float4-threads per token = 4112 slots
  for (int bh = chunk * 4; bh < chunk * 4 + 4; ++bh) {
    const float* xb = src + (size_t)bh * SEQ * D + (size_t)n * TOKENS * D;
    float*       ob = dst + (size_t)bh * SEQ * D + (size_t)n * TOKENS * D;
    for (int slot = threadIdx.x; slot < SLOTS; slot += blockDim.x) {
      int tok = slot >> 4;
      int t   = slot & 15;        // which float4 of the 64-dim head
      size_t off = (size_t)tok * D + t * 4;
      v4f x = __builtin_nontemporal_load((const v4f*)(xb + off));
      v4f r = x;
      if (t < 8) {                // channels < 32: pairs (4t,4t+1),(4t+2,4t+3)
        v4f cp = *(const v4f*)&cs[tok * PSTRIDE + 2 * t]; // {c0,s0,c1,s1}
        r.x = x.x * cp.x - x.y * cp.y;                    // pair 2t  (ch<30 always)
        r.y = x.y * cp.x + x.x * cp.y;
        if (t < 7) {                                      // pair 2t+1 rotated iff ch<30
          r.z = x.z * cp.z - x.w * cp.w;
          r.w = x.w * cp.z + x.z * cp.w;
        }
      }
      __builtin_nontemporal_store(r, (v4f*)(ob + off));
    }
  }
}

// ---------------------------------------------------------------------------
extern "C" void kernel_launch(void* const* d_in, const int* in_sizes, int n_in,
                              void* d_out, int out_size, void* d_ws, size_t ws_size,
                              hipStream_t stream) {
  const float* q     = (const float*)d_in[0];
  const float* k     = (const float*)d_in[1];
  const float* freqs = (const float*)d_in[2];
  float2* tbl = (float2*)d_ws;                 // 1,052,672 bytes used
  float*  out = (float*)d_out;

  const int TOT = NFRAMES * TOKENS * PSTRIDE;  // 131,584 table entries
  build_cs_table<<<(TOT + 255) / 256, 256, 0, stream>>>(freqs, tbl);

  // 2 tensors * 32 frames * 16 chunks = 1024 blocks
  frame_rope_main<<<1024, 256, 0, stream>>>(q, k, tbl, out);
}